// NegSamplingWord2Vec_20160576488011
// MI455X (gfx1250) — compile-verified
//
#include <hip/hip_runtime.h>
#include <math.h>

// Problem constants (match reference)
#define VOCAB   100000
#define DIM     128
#define BATCH   16384
#define OWIN    10          // outside window
#define KNEG    10          // negatives per outside word
#define NSCORE  110         // scores per batch element = OWIN*(KNEG+1)
#define WAVES   8           // waves (batch elements) per 256-thread block
#define ROUNDS  4           // ceil(110/32)

// 16-byte int vector + address-space-qualified variants for the gfx1250
// async global->LDS builtin (param types per hipcc diagnostic: v4i in AS1/AS3).
typedef int v4i __attribute__((ext_vector_type(4)));
typedef __attribute__((address_space(1))) v4i as1_v4i;
typedef __attribute__((address_space(3))) v4i as3_v4i;

// WMMA vector types (16x16x4 f32: A/B = 2 VGPRs, C/D = 8 VGPRs on wave32)
typedef __attribute__((ext_vector_type(2))) float v2f;
typedef __attribute__((ext_vector_type(8))) float v8f;

__device__ __forceinline__ float logsig(float x) {
    // log(sigmoid(x)) = min(x,0) - log1p(exp(-|x|))  (numerically stable)
    return fminf(x, 0.0f) - log1pf(__expf(-fabsf(x)));
}

__global__ __launch_bounds__(256, 4)
void w2v_neg_sampling_loss(const float* __restrict__ center_vectors,
                           const float* __restrict__ outside_vectors,
                           const int*   __restrict__ center_word_index,
                           const int*   __restrict__ outside_word_indices,
                           const int*   __restrict__ negative_samples,
                           float*       __restrict__ losses)
{
    // One center vector Z per wave, staged in LDS (512 B/wave, 4 KB/block).
    __shared__ __align__(16) float zbuf[WAVES][DIM];

    const int lane = threadIdx.x & 31;
    const int wave = threadIdx.x >> 5;
    const int b    = blockIdx.x * WAVES + wave;

    // ---- Stage Z: async global -> LDS (CDNA5 ASYNC path), 16 B per lane ----
    const int ci = center_word_index[b];
    const float* zsrc = center_vectors + (size_t)ci * DIM + 4 * lane;
    float*       zdst = &zbuf[wave][4 * lane];

#if __has_builtin(__builtin_amdgcn_global_load_async_to_lds_b128)
    __builtin_amdgcn_global_load_async_to_lds_b128(
        (as1_v4i*)(unsigned long long)zsrc,
        (as3_v4i*)(unsigned int)(unsigned long long)zdst,
        0, 0);
#if __has_builtin(__builtin_amdgcn_s_wait_asynccnt)
    __builtin_amdgcn_s_wait_asynccnt(0);
#else
    asm volatile("s_wait_asynccnt 0" ::: "memory");
#endif
    asm volatile("" ::: "memory");   // keep LDS reads below the wait
#else
    // Fallback: plain load + LDS store (same-wave producer/consumer, DScnt ordered)
    *(float4*)zdst = *(const float4*)zsrc;
#endif

    // ---- Per-lane score assignment: lane owns score s = r*32 + lane ----
    // s -> (o = s/11, j = s%11); j==0: positive (index owi), j>=1: negative.
    int  vidx[ROUNDS];
    int  isneg[ROUNDS];
    int  msk[ROUNDS];
    #pragma unroll
    for (int r = 0; r < ROUNDS; ++r) {
        const int s     = r * 32 + lane;
        const int valid = (s < NSCORE);
        const int ss    = valid ? s : 0;
        const int o     = ss / 11;
        const int j     = ss - o * 11;

        const int xo = outside_word_indices[b * OWIN + o];
        const int kk = (j > 0) ? (j - 1) : 0;
        const int w  = negative_samples[(b * OWIN + o) * KNEG + kk];

        const int is_pos     = (j == 0);
        const int masked_neg = (xo == 0) || (w == 0);
        const int masked     = is_pos ? (xo == 0) : masked_neg;
        const int idx        = is_pos ? xo : (masked_neg ? 0 : w);

        vidx[r]  = idx;
        isneg[r] = !is_pos;
        msk[r]   = masked || !valid;

        __builtin_prefetch(outside_vectors + (size_t)idx * DIM, 0, 3);
    }

    // ---- Main compute: each lane streams its gathered row vs shared Z ----
    const float4* zrow = (const float4*)zbuf[wave];   // uniform-address LDS broadcast
    float loss = 0.0f;
    #pragma unroll
    for (int r = 0; r < ROUNDS; ++r) {
        const float4* vp = (const float4*)(outside_vectors + (size_t)vidx[r] * DIM);
        float acc = 0.0f;
        #pragma unroll 8
        for (int c = 0; c < DIM / 4; ++c) {
            const float4 v = vp[c];
            const float4 z = zrow[c];
            acc = fmaf(v.x, z.x, acc);
            acc = fmaf(v.y, z.y, acc);
            acc = fmaf(v.z, z.z, acc);
            acc = fmaf(v.w, z.w, acc);
        }
        const float x  = isneg[r] ? -acc : acc;
        const float ls = logsig(x);
        loss += msk[r] ? 0.0f : ls;
    }

    // ---- 32-lane loss reduction via the matrix pipe (ones-matrix trick) ----
    // A row m = {p_m, 0, p_{m+16}, 0} (documented 16x4 f32 A layout), B = ones
    // (all-ones is layout-invariant) => D[m,n] = p_m + p_{m+16} for all n.
    // Summing D's 8 VGPRs gives each half-wave one half of the total
    // (documented 16x16 f32 C/D layout); one shfl_xor(16) completes the sum.
    // EXEC is all-ones here (no divergence yet), as WMMA requires.
    float total;
#if __has_builtin(__builtin_amdgcn_wmma_f32_16x16x4_f32)
    {
        v2f a;   a.x = loss;  a.y = 0.0f;
        v2f one; one.x = 1.0f; one.y = 1.0f;
        v8f d = {};
        d = __builtin_amdgcn_wmma_f32_16x16x4_f32(
                /*neg_a=*/false, a, /*neg_b=*/false, one,
                /*c_mod=*/(short)0, d, /*reuse_a=*/false, /*reuse_b=*/false);
        float h = ((d[0] + d[1]) + (d[2] + d[3])) + ((d[4] + d[5]) + (d[6] + d[7]));
        total = h + __shfl_xor(h, 16, 32);
    }
#else
    total = loss;
    #pragma unroll
    for (int m = 16; m > 0; m >>= 1)
        total += __shfl_xor(total, m, 32);
#endif

    if (lane == 0)
        losses[b] = -total;
}

extern "C" void kernel_launch(void* const* d_in, const int* in_sizes, int n_in,
                              void* d_out, int out_size, void* d_ws, size_t ws_size,
                              hipStream_t stream) {
    (void)in_sizes; (void)n_in; (void)out_size; (void)d_ws; (void)ws_size;
    const float* center_vectors       = (const float*)d_in[0];
    const float* outside_vectors      = (const float*)d_in[1];
    const int*   center_word_index    = (const int*)  d_in[2];
    const int*   outside_word_indices = (const int*)  d_in[3];
    const int*   negative_samples     = (const int*)  d_in[4];
    float*       losses               = (float*)d_out;

    dim3 grid(BATCH / WAVES);   // 2048 blocks
    dim3 block(256);            // 8 wave32 per block, one batch element per wave
    hipLaunchKernelGGL(w2v_neg_sampling_loss, grid, block, 0, stream,
                       center_vectors, outside_vectors, center_word_index,
                       outside_word_indices, negative_samples, losses);
}